// MappedTransposedConvolution_34282428956675
// MI455X (gfx1250) — compile-verified
//
#include <hip/hip_runtime.h>
#include <hip/hip_bf16.h>

typedef __attribute__((ext_vector_type(2))) float v2f;
typedef __attribute__((ext_vector_type(8))) float v8f;

#define CIN   64
#define COUT  64
#define KTAPS 9
#define H_IN  256
#define W_IN  512
#define PTOT  (H_IN * W_IN)   // 131072 input pixels
#define OH    512
#define OW    1024
#define OHOW  (OH * OW)       // 524288

// ---------------------------------------------------------------------------
// Kernel A: repack weight [Cin,Cout,K] into per-lane B-fragment order for
// v_wmma_f32_16x16x4_f32.  B (4x16, KxN) layout: lane l holds n = l%16;
// VGPR0 -> k = 2*(l/16), VGPR1 -> k = 2*(l/16)+1.  Packed scalar index:
//   ((((wv*9 + k)*16 + kk)*32 + lane)*2 + j)  ->  weight[(c*64 + o)*9 + k]
//   with c = 4*kk + 2*(lane/16) + j,  o = 16*wv + (lane&15).
// Total 36864 floats = 147456 bytes in d_ws; lane-contiguous float2 loads.
// ---------------------------------------------------------------------------
__global__ __launch_bounds__(256) void repack_weight_kernel(
    const float* __restrict__ weight, float* __restrict__ packed) {
  int t = blockIdx.x * 256 + threadIdx.x;          // 0 .. 36863
  int j    = t & 1;
  int lane = (t >> 1) & 31;
  int kk   = (t >> 6) & 15;
  int rest = t >> 10;                              // wv*9 + k, 0..35
  int k    = rest % KTAPS;
  int wv   = rest / KTAPS;
  int c = 4 * kk + 2 * (lane >> 4) + j;
  int o = 16 * wv + (lane & 15);
  packed[t] = weight[(c * COUT + o) * KTAPS + k];
}

// ---------------------------------------------------------------------------
// Kernel B: out[o, q] = bias[o]   (out is [64, 524288] f32; OHOW = 1<<19)
// ---------------------------------------------------------------------------
__global__ __launch_bounds__(256) void bias_init_kernel(
    const float* __restrict__ bias, float* __restrict__ out) {
  size_t i = (size_t)blockIdx.x * 256 + threadIdx.x;
  out[i] = bias[i >> 19];
}

// ---------------------------------------------------------------------------
// Kernel C: fused GEMM (WMMA f32 16x16x4) + bilinear scatter-splat.
// 1 block = 16 pixels, 128 threads = 4 waves; wave wv owns channels
// [16*wv, 16*wv+16).  Output (134 MB) stays resident in the 192 MB L2, so
// the 302M global_atomic_add_f32 resolve in L2.
// ---------------------------------------------------------------------------
__global__ __launch_bounds__(128) void mapped_tconv_kernel(
    const float* __restrict__ x,        // [Cin, P]
    const float* __restrict__ packedW,  // repacked B fragments (d_ws)
    const float* __restrict__ smap,     // [P, K, 2] (x, y)
    float* __restrict__ out) {          // [Cout, OH*OW]
  __shared__ float xtile[CIN * 16];     // [c][m] : 4 KB
  __shared__ int   cidx[16 * 4];        // per-pixel 4 corner indices
  __shared__ float cwt[16 * 4];         // per-pixel 4 corner weights

  const int t    = threadIdx.x;
  const int lane = t & 31;
  const int wv   = t >> 5;
  const int p0   = blockIdx.x * 16;

  // ---- load x tile (64 cin x 16 pixels) into LDS, reused for all 9 taps ----
#pragma unroll
  for (int i = 0; i < 8; ++i) {
    int lin = t + 128 * i;
    int c = lin >> 4, m = lin & 15;
    xtile[lin] = x[(size_t)c * PTOT + p0 + m];
  }
  __syncthreads();

  // ---- A fragments in registers (independent of tap k) ----
  // A (16x4 f32): lane l -> M = l%16; VGPR0: K = 2*(l/16), VGPR1: K = 2*(l/16)+1
  float a0[16], a1[16];
  const int m_a   = lane & 15;
  const int chalf = (lane >> 4) << 1;  // 0 or 2
#pragma unroll
  for (int kk = 0; kk < 16; ++kk) {
    int c0 = 4 * kk + chalf;
    a0[kk] = xtile[c0 * 16 + m_a];
    a1[kk] = xtile[(c0 + 1) * 16 + m_a];
  }

  const float2* pw = (const float2*)packedW;
  const int m_s    = (lane >> 4) ? 8 : 0;            // D-row half offset
  const int nglob  = 16 * wv + (lane & 15);          // this lane's out channel
  float* outn      = out + (size_t)nglob * OHOW;

  for (int k = 0; k < KTAPS; ++k) {
    __syncthreads();  // protect cidx/cwt from previous iteration's readers
    // ---- 64 threads: bilinear corner indices + weights for 16 pixels ----
    if (t < 64) {
      int m = t >> 2, cc = t & 3;
      int p = p0 + m;
      float sx  = smap[(p * KTAPS + k) * 2 + 0];
      float sy  = smap[(p * KTAPS + k) * 2 + 1];
      float x0f = floorf(sx), y0f = floorf(sy);
      float dx = sx - x0f, dy = sy - y0f;
      int xi = (int)x0f + (cc & 1);
      int yi = (int)y0f + (cc >> 1);
      float wt = ((cc & 1) ? dx : 1.0f - dx) * ((cc >> 1) ? dy : 1.0f - dy);
      bool valid = (xi >= 0) & (xi < OW) & (yi >= 0) & (yi < OH);
      if (!valid) wt = 0.0f;               // match reference: zero weight OOB
      int xc = min(max(xi, 0), OW - 1);
      int yc = min(max(yi, 0), OH - 1);
      cidx[t] = yc * OW + xc;
      cwt[t]  = wt;
    }
    __syncthreads();

    // ---- GEMM: 16x16 tap tile, K=Cin=64 via 16 x (16x16x4 f32 WMMA) ----
    v8f d = {0.f, 0.f, 0.f, 0.f, 0.f, 0.f, 0.f, 0.f};
#pragma unroll
    for (int kk = 0; kk < 16; ++kk) {
      v2f a = {a0[kk], a1[kk]};
      float2 bw = pw[((wv * KTAPS + k) * 16 + kk) * 32 + lane];
      v2f b = {bw.x, bw.y};
      d = __builtin_amdgcn_wmma_f32_16x16x4_f32(
          /*neg_a=*/false, a, /*neg_b=*/false, b,
          /*c_mod=*/(short)0, d, /*reuse_a=*/false, /*reuse_b=*/false);
    }

    // ---- scatter: 8 pixels/lane x 4 corners, L2-resident f32 atomics ----
#pragma unroll
    for (int v = 0; v < 8; ++v) {
      int m = v + m_s;
      float val = d[v];
#pragma unroll
      for (int cc = 0; cc < 4; ++cc) {
        unsafeAtomicAdd(&outn[cidx[m * 4 + cc]], val * cwt[m * 4 + cc]);
      }
    }
  }
}

extern "C" void kernel_launch(void* const* d_in, const int* in_sizes, int n_in,
                              void* d_out, int out_size, void* d_ws, size_t ws_size,
                              hipStream_t stream) {
  const float* x      = (const float*)d_in[0];  // [1,64,256,512]
  const float* weight = (const float*)d_in[1];  // [64,64,9]
  const float* bias   = (const float*)d_in[2];  // [64]
  const float* smap   = (const float*)d_in[3];  // [256,512,9,2]
  float* out          = (float*)d_out;          // [1,64,512,1024]
  float* packedW      = (float*)d_ws;           // 147456 bytes

  // A: repack weights into WMMA B-fragment order (36864 elems, 144 blocks)
  repack_weight_kernel<<<(CIN * COUT * KTAPS) / 256, 256, 0, stream>>>(weight, packedW);

  // B: out = bias broadcast (64 * 524288 elems)
  bias_init_kernel<<<(COUT * OHOW) / 256, 256, 0, stream>>>(bias, out);

  // C: fused WMMA GEMM + bilinear scatter (8192 blocks x 128 threads)
  mapped_tconv_kernel<<<PTOT / 16, 128, 0, stream>>>(x, packedW, smap, out);
}